// GCN_18889266168413
// MI455X (gfx1250) — compile-verified
//
#include <hip/hip_runtime.h>

typedef __attribute__((ext_vector_type(2))) float v2f;
typedef __attribute__((ext_vector_type(8))) float v8f;

#define N_NODES 100000
#define N_EDGES 1000000
#define IN_CH   128
#define HID_CH  64
#define OUT_CH  64

// ---------------- degree / symmetric norm ----------------
__global__ void k_init_deg(float* deg, int n) {
    int i = blockIdx.x * blockDim.x + threadIdx.x;
    if (i < n) deg[i] = 1.0f;                    // self-loop contributes 1
}

__global__ void k_deg_accum(const int* __restrict__ dst, float* deg, int e) {
    int i = blockIdx.x * blockDim.x + threadIdx.x;
    if (i < e) atomicAdd(&deg[dst[i]], 1.0f);
}

__global__ void k_dinv(float* deg, int n) {
    int i = blockIdx.x * blockDim.x + threadIdx.x;
    if (i < n) deg[i] = rsqrtf(deg[i]);          // deg >= 1 always
}

// ---------------- seed output with bias ----------------
__global__ void k_init_bias(float* __restrict__ out, const float* __restrict__ b, int total) {
    int i = blockIdx.x * blockDim.x + threadIdx.x;
    if (i < total) out[i] = b[i & 63];
}

// ---------------- ReLU ----------------
__global__ void k_relu(float* a, int n) {
    int i = blockIdx.x * blockDim.x + threadIdx.x;
    if (i < n) a[i] = fmaxf(a[i], 0.0f);
}

// ---------------- GEMM: H[N x 64] = X[N x K] * W[K x 64] via f32 WMMA ----------------
// One wave (32 lanes) computes a 16-row x 64-col output tile.
// A 16x4 f32 layout: lane = M (mod 16); lanes 0-15 hold K pair {k0,k0+1},
//                    lanes 16-31 hold {k0+2,k0+3}  (v0=first, v1=second).
// B 4x16 f32 layout: lane = N (mod 16); same K-pair split across half-waves.
// C/D 16x16 f32:     VGPR r -> M=r (lanes 0-15) / M=8+r (lanes 16-31), lane = N.
//
// W is staged into LDS in K-pair-interleaved form so each B operand is one
// contiguous 8-byte v2f:  sW[(k/2)*128 + c*2 + (k&1)] = W[k][c].
// The 4 N-tiles then sit at +0/+128/+256/+384 bytes from a single lane address
// -> two ds_load_2addr_b64 per K-step, no register shuffles.
__global__ void k_gemm_wmma(const float* __restrict__ X, const float* __restrict__ W,
                            float* __restrict__ H, int nTiles, int K) {
    extern __shared__ float sW[];                // K x 64 floats, interleaved
    const int tid = threadIdx.x;
    for (int i = tid; i < K * 64; i += blockDim.x) {
        const int k = i >> 6;                    // coalesced read of W[k][c]
        const int c = i & 63;
        sW[(k >> 1) * 128 + c * 2 + (k & 1)] = W[i];
    }
    __syncthreads();

    const int wave = tid >> 5;
    const int lane = tid & 31;
    const int tile = blockIdx.x * (blockDim.x >> 5) + wave;
    if (tile >= nTiles) return;                  // wave-uniform: EXEC stays all-1s

    const int laneM = lane & 15;
    const int hi    = lane >> 4;                 // half-wave selects K sub-pair

    const float* __restrict__ xrow = X + (size_t)(tile * 16 + laneM) * K + hi * 2;
    const v2f*   __restrict__ wlane = (const v2f*)sW + hi * 64 + laneM;

    v8f acc0 = {}, acc1 = {}, acc2 = {}, acc3 = {};

    for (int k0 = 0; k0 < K; k0 += 4) {
        v2f a = *(const v2f*)(xrow + k0);        // A pair {k0(+hi*2), k0+1(+hi*2)}

        const v2f* w = wlane + (k0 >> 1) * 64;   // pair-row (k0/2 + hi), col laneM
        v2f b0 = w[0];                           // N-tile 0: col laneM
        v2f b1 = w[16];                          // N-tile 1: col laneM+16
        v2f b2 = w[32];                          // N-tile 2: col laneM+32
        v2f b3 = w[48];                          // N-tile 3: col laneM+48

        acc0 = __builtin_amdgcn_wmma_f32_16x16x4_f32(false, a, false, b0, (short)0, acc0, false, false);
        acc1 = __builtin_amdgcn_wmma_f32_16x16x4_f32(false, a, false, b1, (short)0, acc1, false, false);
        acc2 = __builtin_amdgcn_wmma_f32_16x16x4_f32(false, a, false, b2, (short)0, acc2, false, false);
        acc3 = __builtin_amdgcn_wmma_f32_16x16x4_f32(false, a, false, b3, (short)0, acc3, false, false);
    }

    // Store: lanes 0-15 cover M=0..7 (VGPR r -> M=r), lanes 16-31 cover M=8..15.
    float* __restrict__ o = H + (size_t)(tile * 16 + hi * 8) * 64 + laneM;
    #pragma unroll
    for (int r = 0; r < 8; ++r) {
        o[r * 64 + 0]  = acc0[r];
        o[r * 64 + 16] = acc1[r];
        o[r * 64 + 32] = acc2[r];
        o[r * 64 + 48] = acc3[r];
    }
}

// ---------------- edge scatter: one wave per edge, 2 channels per lane ----------------
__global__ void k_scatter(const float* __restrict__ H, const int* __restrict__ src,
                          const int* __restrict__ dst, const float* __restrict__ dinv,
                          float* __restrict__ out, int nEdges, int nNodes) {
    const long long g = (long long)blockIdx.x * blockDim.x + threadIdx.x;
    const int e    = (int)(g >> 5);
    const int lane = (int)(g & 31);
    const int total = nEdges + nNodes;
    if (e >= total) return;

    int s, d;
    if (e < nEdges) { s = src[e]; d = dst[e]; }
    else            { s = d = e - nEdges; }      // fused self-loops

    const float nrm = dinv[s] * dinv[d];
    const float* hs = H + (size_t)s * 64 + lane * 2;
    const float m0 = hs[0] * nrm;
    const float m1 = hs[1] * nrm;

    float* o = out + (size_t)d * 64 + lane * 2;
    atomicAdd(o,     m0);
    atomicAdd(o + 1, m1);
}

extern "C" void kernel_launch(void* const* d_in, const int* in_sizes, int n_in,
                              void* d_out, int out_size, void* d_ws, size_t ws_size,
                              hipStream_t stream) {
    const float* x   = (const float*)d_in[0];
    const int*   src = (const int*)d_in[1];           // edge_index[0], flat [2, E]
    const int*   dst = src + N_EDGES;                 // edge_index[1]
    const float* W1  = (const float*)d_in[2];
    const float* b1  = (const float*)d_in[3];
    const float* W2  = (const float*)d_in[4];
    const float* b2  = (const float*)d_in[5];
    float* out = (float*)d_out;

    // workspace carve-up
    float* dinv = (float*)d_ws;                                        // 0.4 MB
    float* h    = (float*)((char*)d_ws + (size_t)512 * 1024);          // 25.6 MB
    float* agg  = (float*)((char*)d_ws + (size_t)512 * 1024
                                       + (size_t)26 * 1024 * 1024);    // 25.6 MB

    const int nElems = N_NODES * 64;                  // 6.4M
    const int nTiles = N_NODES / 16;                  // 6250 (exact)
    const int gemmBlocks = (nTiles + 7) / 8;          // 8 waves/block
    const int eTotal = N_EDGES + N_NODES;
    const long long scatterThreads = (long long)eTotal * 32;
    const int scatterBlocks = (int)((scatterThreads + 255) / 256);

    // norm = deg^{-1/2}
    k_init_deg <<<(N_NODES + 255) / 256, 256, 0, stream>>>(dinv, N_NODES);
    k_deg_accum<<<(N_EDGES + 255) / 256, 256, 0, stream>>>(dst, dinv, N_EDGES);
    k_dinv     <<<(N_NODES + 255) / 256, 256, 0, stream>>>(dinv, N_NODES);

    // ---- layer 1 ----
    k_gemm_wmma<<<gemmBlocks, 256, IN_CH * 64 * sizeof(float), stream>>>(x, W1, h, nTiles, IN_CH);
    k_init_bias<<<(nElems + 255) / 256, 256, 0, stream>>>(agg, b1, nElems);
    k_scatter  <<<scatterBlocks, 256, 0, stream>>>(h, src, dst, dinv, agg, N_EDGES, N_NODES);
    k_relu     <<<(nElems + 255) / 256, 256, 0, stream>>>(agg, nElems);

    // ---- layer 2 ----
    k_gemm_wmma<<<gemmBlocks, 256, HID_CH * 64 * sizeof(float), stream>>>(agg, W2, h, nTiles, HID_CH);
    k_init_bias<<<(nElems + 255) / 256, 256, 0, stream>>>(out, b2, nElems);
    k_scatter  <<<scatterBlocks, 256, 0, stream>>>(h, src, dst, dinv, out, N_EDGES, N_NODES);
}